// Model_33174327395032
// MI455X (gfx1250) — compile-verified
//
#include <hip/hip_runtime.h>
#include <stdint.h>

#define BB 8
#define NN 4096
#define CC 128
#define GG 1024
#define KK 32
#define CH 131      // C+3
#define OC 259      // 2C+3
#define EPSF 1e-5f

typedef float v2f __attribute__((ext_vector_type(2)));
typedef float v8f __attribute__((ext_vector_type(8)));

__device__ __forceinline__ unsigned long long u64max(unsigned long long a, unsigned long long b) { return a > b ? a : b; }
__device__ __forceinline__ unsigned long long u64min(unsigned long long a, unsigned long long b) { return a < b ? a : b; }

// Full 32-lane bitonic sort, ascending by 64-bit key (dist_bits<<32 | index).
__device__ __forceinline__ unsigned long long bsort32(unsigned long long key, int lane) {
  #pragma unroll
  for (int k2 = 2; k2 <= 32; k2 <<= 1) {
    #pragma unroll
    for (int j = k2 >> 1; j >= 1; j >>= 1) {
      unsigned long long partner = __shfl_xor(key, j, 32);
      bool up  = ((lane & k2) == 0);
      bool low = ((lane & j)  == 0);
      unsigned long long mn = u64min(key, partner);
      unsigned long long mx = u64max(key, partner);
      key = (up == low) ? mn : mx;
    }
  }
  return key;
}

// Bitonic "clean" merge: input bitonic -> ascending sorted.
__device__ __forceinline__ unsigned long long bmerge32(unsigned long long key, int lane) {
  #pragma unroll
  for (int j = 16; j >= 1; j >>= 1) {
    unsigned long long partner = __shfl_xor(key, j, 32);
    unsigned long long mn = u64min(key, partner);
    unsigned long long mx = u64max(key, partner);
    key = ((lane & j) == 0) ? mn : mx;
  }
  return key;
}

// ---------------------------------------------------------------------------
// Kernel 0: zero the per-batch sum-of-squares accumulators.
// ---------------------------------------------------------------------------
__global__ void pg_init(float* ss) {
  if (threadIdx.x < BB) ss[threadIdx.x] = 0.0f;
}

// ---------------------------------------------------------------------------
// Kernel 1: Farthest Point Sampling. One workgroup per batch (256 thr, 8 waves).
// dist_min lives in registers (16 points/thread); argmax via wave shfl + LDS.
// Also emits new_xyz (to d_out and workspace for the KNN A-operand).
// ---------------------------------------------------------------------------
__global__ void pg_fps(const float* __restrict__ xyz, int* __restrict__ fps_idx,
                       float* __restrict__ nxyz_ws, float* __restrict__ out_xyz) {
  const int b = blockIdx.x, tid = threadIdx.x;
  const int lane = tid & 31, wid = tid >> 5;
  __shared__ float cent[3];
  __shared__ unsigned long long wkey[8];
  __shared__ int s_win;

  const float* base = xyz + (size_t)b * NN * 3;
  float px[16], py[16], pz[16], dm[16];
  #pragma unroll
  for (int i = 0; i < 16; i++) {
    int n = tid + i * 256;
    px[i] = base[n * 3 + 0];
    py[i] = base[n * 3 + 1];
    pz[i] = base[n * 3 + 2];
    dm[i] = 1e10f;
  }
  if (tid == 0) s_win = 0;
  __syncthreads();

  for (int g = 0; g < GG; g++) {
    if (tid == 0) {
      int cur = s_win;
      fps_idx[b * GG + g] = cur;
      float cx = base[cur * 3 + 0], cy = base[cur * 3 + 1], cz = base[cur * 3 + 2];
      cent[0] = cx; cent[1] = cy; cent[2] = cz;
      size_t o = ((size_t)b * GG + g) * 3;
      nxyz_ws[o + 0] = cx; nxyz_ws[o + 1] = cy; nxyz_ws[o + 2] = cz;
      out_xyz[o + 0] = cx; out_xyz[o + 1] = cy; out_xyz[o + 2] = cz;
    }
    __syncthreads();
    const float cx = cent[0], cy = cent[1], cz = cent[2];
    unsigned long long k = 0ull;
    #pragma unroll
    for (int i = 0; i < 16; i++) {
      float dx = px[i] - cx, dy = py[i] - cy, dz = pz[i] - cz;
      float d = dx * dx + dy * dy + dz * dz;
      dm[i] = fminf(dm[i], d);
      // max over (dist, then LOWEST index on ties) -> pack inverted index.
      unsigned long long kk =
          ((unsigned long long)__float_as_uint(dm[i]) << 32) |
          (unsigned)(0xFFFFFFFFu - (unsigned)(tid + i * 256));
      k = u64max(k, kk);
    }
    #pragma unroll
    for (int s = 16; s >= 1; s >>= 1) k = u64max(k, __shfl_xor(k, s, 32));
    if (lane == 0) wkey[wid] = k;
    __syncthreads();
    if (tid == 0) {
      unsigned long long m = wkey[0];
      #pragma unroll
      for (int w = 1; w < 8; w++) m = u64max(m, wkey[w]);
      s_win = (int)(0xFFFFFFFFu - (unsigned)(m & 0xFFFFFFFFull));
    }
    __syncthreads();
  }
}

// ---------------------------------------------------------------------------
// Kernel 2: KNN. One block per (batch, 16-group tile); 256 thr = 8 waves.
// Distances for a 16x512 chunk computed with V_WMMA_F32_16X16X4_F32:
//   A row m = (-2sx, -2sy, -2sz, |s|^2),  B col n = (px, py, pz, 1)
//   => C = -2 s.p + |s|^2 ; add |p|^2 per column.
// Top-32 per row maintained as an ascending sorted set across the 32 lanes,
// merged per 32-candidate batch with bitonic sort + bitonic merge.
// ---------------------------------------------------------------------------
__global__ void pg_knn(const float* __restrict__ xyz, const float* __restrict__ nxyz,
                       int* __restrict__ knn_idx) {
  __shared__ float dist_s[16 * 516];   // 16 rows, 512 cols + pad (33 KB LDS)
  const int blk = blockIdx.x;          // 0..511
  const int b = blk >> 6;
  const int gbase = (blk & 63) * 16;
  const int tid = threadIdx.x, lane = tid & 31, wid = tid >> 5;
  const int hl = lane >> 4, m = lane & 15;
  const float* xb = xyz + (size_t)b * NN * 3;

  // Build the A operand (16x4 f32: lanes 0-15 hold K0,K1; lanes 16-31 K2,K3).
  const float* sp = nxyz + ((size_t)b * GG + gbase + m) * 3;
  float sx = sp[0], sy = sp[1], sz = sp[2];
  float s2 = sx * sx + sy * sy + sz * sz;
  v2f a;
  a.x = hl ? (-2.0f * sz) : (-2.0f * sx);
  a.y = hl ? s2           : (-2.0f * sy);

  // Running top-32 per row: lane l holds slot l, ascending sorted keys.
  unsigned long long cur0 = (0x7F800000ull << 32) | (unsigned)lane;  // +inf
  unsigned long long cur1 = cur0;

  for (int chk = 0; chk < 8; chk++) {
    const int cb = chk * 512;
    if (chk < 7) __builtin_prefetch(xb + (size_t)(cb + 512) * 3 + tid * 6, 0, 1);

    // ---- compute phase: each wave produces 4 WMMA column tiles ----
    #pragma unroll
    for (int t = 0; t < 4; t++) {
      const int tile = wid * 4 + t;
      const int n = cb + tile * 16 + m;
      float px = xb[n * 3 + 0], py = xb[n * 3 + 1], pz = xb[n * 3 + 2];
      float dd = px * px + py * py + pz * pz;
      v2f bv;                 // B 4x16: lanes 0-15 hold K0,K1; lanes 16-31 K2,K3
      bv.x = hl ? pz : px;
      bv.y = hl ? 1.0f : py;
      v8f c = {};
      c = __builtin_amdgcn_wmma_f32_16x16x4_f32(false, a, false, bv,
                                                (short)0, c, false, false);
      const int col = tile * 16 + m;
      #pragma unroll
      for (int r = 0; r < 8; r++) {
        int row = r + 8 * hl;                       // C layout: M = r + 8*(l>>4)
        dist_s[row * 516 + col] = fmaxf(c[r] + dd, 0.0f);
      }
    }
    __syncthreads();

    // ---- selection phase: wave w owns rows 2w and 2w+1 ----
    #pragma unroll
    for (int rr = 0; rr < 2; rr++) {
      const int row = wid * 2 + rr;
      unsigned long long cur = rr ? cur1 : cur0;
      for (int i = 0; i < 16; i++) {
        const int cidx = i * 32 + lane;
        float d = dist_s[row * 516 + cidx];
        unsigned long long key =
            ((unsigned long long)__float_as_uint(d) << 32) | (unsigned)(cb + cidx);
        key = bsort32(key, lane);             // ascending batch
        key = __shfl_xor(key, 31, 32);        // -> descending
        cur = u64min(cur, key);               // bitonic half-merge keeps 32 smallest
        cur = bmerge32(cur, lane);            // restore ascending sorted
      }
      if (rr) cur1 = cur; else cur0 = cur;
    }
    __syncthreads();
  }

  knn_idx[((size_t)b * GG + gbase + wid * 2 + 0) * KK + lane] = (int)(cur0 & 0xFFFFFFFFull);
  knn_idx[((size_t)b * GG + gbase + wid * 2 + 1) * KK + lane] = (int)(cur1 & 0xFFFFFFFFull);
}

// ---------------------------------------------------------------------------
// Kernel 3: per-(b,g,ch) mean over K and per-batch centered sum-of-squares.
// One block per (b,g); thread = channel (131 of 256 active for gathers).
// ---------------------------------------------------------------------------
__global__ void pg_stats(const float* __restrict__ xyz, const float* __restrict__ points,
                         const int* __restrict__ knn_idx, float* __restrict__ mean_ws,
                         float* __restrict__ ss_ws) {
  const int blk = blockIdx.x, b = blk >> 10, tid = threadIdx.x;
  __shared__ int sid[KK];
  __shared__ float red[256];
  if (tid < KK) sid[tid] = knn_idx[(size_t)blk * KK + tid];
  __syncthreads();

  float ss = 0.0f;
  if (tid < CH) {
    float v[KK];
    float sum = 0.0f;
    #pragma unroll
    for (int k = 0; k < KK; k++) {
      int id = sid[k];
      float x = (tid < CC) ? points[((size_t)b * NN + id) * CC + tid]
                           : xyz[((size_t)b * NN + id) * 3 + (tid - CC)];
      v[k] = x;
      sum += x;
    }
    float mean = sum * (1.0f / KK);
    mean_ws[(size_t)blk * CH + tid] = mean;
    #pragma unroll
    for (int k = 0; k < KK; k++) { float d = v[k] - mean; ss += d * d; }
  }
  red[tid] = ss;
  __syncthreads();
  for (int s = 128; s >= 1; s >>= 1) {
    if (tid < s) red[tid] += red[tid + s];
    __syncthreads();
  }
  if (tid == 0) atomicAdd(&ss_ws[b], red[0]);
}

// ---------------------------------------------------------------------------
// Kernel 4: finalize per-batch std (ddof=1).
// ---------------------------------------------------------------------------
__global__ void pg_std(const float* __restrict__ ss, float* __restrict__ stdv) {
  int t = threadIdx.x;
  if (t < BB) {
    float n = (float)GG * (float)KK * (float)CH - 1.0f;
    stdv[t] = sqrtf(ss[t] / n);
  }
}

// ---------------------------------------------------------------------------
// Kernel 5: fused normalize + affine + anchor-concat, streaming 272 MB out.
// One block per (b,g); coalesced writes over [K,259].
// ---------------------------------------------------------------------------
__global__ void pg_out(const float* __restrict__ xyz, const float* __restrict__ points,
                       const float* __restrict__ alpha, const float* __restrict__ beta,
                       const int* __restrict__ knn_idx, const int* __restrict__ fps_idx,
                       const float* __restrict__ mean_ws, const float* __restrict__ stdv,
                       float* __restrict__ out_pts) {
  const int blk = blockIdx.x, b = blk >> 10, tid = threadIdx.x;
  __shared__ int sid[KK];
  __shared__ float sm[CH], sa[CH], sb[CH], sc[CC];
  if (tid < KK) sid[tid] = knn_idx[(size_t)blk * KK + tid];
  if (tid < CH) {
    sm[tid] = mean_ws[(size_t)blk * CH + tid];
    sa[tid] = alpha[tid];
    sb[tid] = beta[tid];
  }
  if (tid < CC) {
    int cidx = fps_idx[blk];
    sc[tid] = points[((size_t)b * NN + cidx) * CC + tid];
  }
  __syncthreads();
  const float inv = 1.0f / (stdv[b] + EPSF);
  float* ob = out_pts + (size_t)blk * KK * OC;
  for (int t = tid; t < KK * OC; t += 256) {
    int k = t / OC, ch = t - k * OC;
    float w;
    if (ch < CH) {
      int id = sid[k];
      float x = (ch < CC) ? points[((size_t)b * NN + id) * CC + ch]
                          : xyz[((size_t)b * NN + id) * 3 + (ch - CC)];
      w = sa[ch] * (x - sm[ch]) * inv + sb[ch];
    } else {
      w = sc[ch - CH];
    }
    ob[t] = w;
  }
}

// ---------------------------------------------------------------------------
extern "C" void kernel_launch(void* const* d_in, const int* in_sizes, int n_in,
                              void* d_out, int out_size, void* d_ws, size_t ws_size,
                              hipStream_t stream) {
  (void)in_sizes; (void)n_in; (void)out_size; (void)ws_size;
  const float* xyz    = (const float*)d_in[0];   // [8,4096,3]
  const float* points = (const float*)d_in[1];   // [8,4096,128]
  const float* alpha  = (const float*)d_in[2];   // [131]
  const float* beta   = (const float*)d_in[3];   // [131]

  float* out_xyz = (float*)d_out;                      // [8,1024,3]
  float* out_pts = (float*)d_out + (size_t)BB * GG * 3; // [8,1024,32,259]

  char* w = (char*)d_ws;
  int*   fps   = (int*)(w + 0);           //  8192 ints   (32 KB)
  float* nxyz  = (float*)(w + 32768);     // 24576 floats (96 KB)
  int*   knn   = (int*)(w + 131072);      // 262144 ints  (1 MB)
  float* meanw = (float*)(w + 1179648);   // 8192*131 floats (4.1 MB)
  float* ssw   = (float*)(w + 5472256);   // 8 floats
  float* stdw  = (float*)(w + 5472320);   // 8 floats

  pg_init<<<1, 64, 0, stream>>>(ssw);
  pg_fps<<<BB, 256, 0, stream>>>(xyz, fps, nxyz, out_xyz);
  pg_knn<<<BB * (GG / 16), 256, 0, stream>>>(xyz, nxyz, knn);
  pg_stats<<<BB * GG, 256, 0, stream>>>(xyz, points, knn, meanw, ssw);
  pg_std<<<1, 32, 0, stream>>>(ssw, stdw);
  pg_out<<<BB * GG, 256, 0, stream>>>(xyz, points, alpha, beta, knn, fps,
                                      meanw, stdw, out_pts);
}